// GumbelSoftmax_38053410243256
// MI455X (gfx1250) — compile-verified
//
#include <hip/hip_runtime.h>
#include <hip/hip_bf16.h>
#include <math.h>

#define GEPS 1e-10f

// Native clang vector type so __builtin_nontemporal_load/store accept it
// (HIP's float4 is a struct and is rejected by the builtin).
typedef float v4f __attribute__((ext_vector_type(4)));

// One block per row. Phase 1: streaming scan of logits+u computing
// z = logits - log(EPS - log(u+EPS)) and the row argmax (lowest index on
// ties, matching jnp.argmax; softmax is monotone so argmax(softmax(z)) ==
// argmax(z)). Phase 2: stream a one-hot row of float32.
// All global traffic is 128-bit vectorized and non-temporal (data is
// touched exactly once -> don't pollute L2/WGP$).
__global__ __launch_bounds__(256) void gumbel_argmax_onehot_kernel(
    const float* __restrict__ logits,
    const float* __restrict__ u,
    float* __restrict__ out,
    int nCols)                      // must be a multiple of 4 (32000 is)
{
    const int row  = blockIdx.x;
    const int tid  = threadIdx.x;
    const int nVec = nCols >> 2;

    const v4f* __restrict__ lg =
        reinterpret_cast<const v4f*>(logits + (size_t)row * nCols);
    const v4f* __restrict__ uu =
        reinterpret_cast<const v4f*>(u + (size_t)row * nCols);

    // ---------------- Phase 1: per-thread argmax over strided float4s ------
    float bestVal = -INFINITY;
    int   bestIdx = 0;

    // Prefetch distance: 8 block-strides ahead = 8*256*16B = 32 KB per stream.
    const int PFD = 8 * 256;

    for (int v = tid; v < nVec; v += 256) {
        if (v + PFD < nVec) {
            __builtin_prefetch(&lg[v + PFD], 0, 0);   // -> global_prefetch_b8
            __builtin_prefetch(&uu[v + PFD], 0, 0);
        }
        const v4f l = __builtin_nontemporal_load(&lg[v]);  // b128, TH_NT
        const v4f r = __builtin_nontemporal_load(&uu[v]);

        // Gumbel noise exactly as the reference formula (float32 throughout).
        const float z0 = l.x - logf(GEPS - logf(r.x + GEPS));
        const float z1 = l.y - logf(GEPS - logf(r.y + GEPS));
        const float z2 = l.z - logf(GEPS - logf(r.z + GEPS));
        const float z3 = l.w - logf(GEPS - logf(r.w + GEPS));

        // In-order lane compares with strict '>' keep the lowest index on ties.
        const int base = v << 2;
        float bv = z0; int bi = base;
        if (z1 > bv) { bv = z1; bi = base + 1; }
        if (z2 > bv) { bv = z2; bi = base + 2; }
        if (z3 > bv) { bv = z3; bi = base + 3; }
        // Indices grow monotonically within a thread, so strict '>' suffices.
        if (bv > bestVal) { bestVal = bv; bestIdx = bi; }
    }

    // ---------------- Block reduction: (max value, min index on ties) ------
    __shared__ float sVal[256];
    __shared__ int   sIdx[256];
    sVal[tid] = bestVal;
    sIdx[tid] = bestIdx;
    __syncthreads();

    #pragma unroll
    for (int s = 128; s > 0; s >>= 1) {
        if (tid < s) {
            const float v2 = sVal[tid + s];
            const int   i2 = sIdx[tid + s];
            if (v2 > sVal[tid] || (v2 == sVal[tid] && i2 < sIdx[tid])) {
                sVal[tid] = v2;
                sIdx[tid] = i2;
            }
        }
        __syncthreads();
    }
    const int arg    = sIdx[0];       // safe: last __syncthreads above
    const int argVec = arg >> 2;
    const int argLn  = arg & 3;

    // ---------------- Phase 2: stream one-hot row (b128 NT stores) ---------
    v4f* __restrict__ o =
        reinterpret_cast<v4f*>(out + (size_t)row * nCols);

    for (int v = tid; v < nVec; v += 256) {
        v4f w = (v4f)(0.0f);
        if (v == argVec) {
            w[argLn] = 1.0f;
        }
        __builtin_nontemporal_store(w, &o[v]);
    }
}

extern "C" void kernel_launch(void* const* d_in, const int* in_sizes, int n_in,
                              void* d_out, int out_size, void* d_ws, size_t ws_size,
                              hipStream_t stream) {
    (void)n_in; (void)d_ws; (void)ws_size;

    const float* logits = (const float*)d_in[0];
    const float* u      = (const float*)d_in[1];
    float*       out    = (float*)d_out;

    const int C = 32000;                 // class dimension from the reference
    const int B = in_sizes[0] / C;       // 4096 rows
    (void)out_size;

    gumbel_argmax_onehot_kernel<<<B, 256, 0, stream>>>(logits, u, out, C);
}